// RollingAttentionPooling_85607288144244
// MI455X (gfx1250) — compile-verified
//
#include <hip/hip_runtime.h>
#include <hip/hip_bf16.h>
#include <math.h>

// Problem constants (from reference)
constexpr int kB   = 8;
constexpr int kT   = 4096;
constexpr int kD   = 1024;
constexpr int kH   = 8;
constexpr int kHD  = 128;
constexpr int kP   = 1024;   // H*HD
constexpr int kOUT = 1024;
constexpr int kSTR = 32;
constexpr int kW   = 128;    // number of windows = T/STRIDE
constexpr int kKC  = 8;      // K-split chunks for K3
constexpr int kKCL = kT / kKC; // 512 rows per chunk

typedef float v2f __attribute__((ext_vector_type(2)));
typedef float v8f __attribute__((ext_vector_type(8)));

// ---------------------------------------------------------------------------
// K0: q_proj[c, h] = sum_dh w_kv[c, h*128+dh] * query[h, dh]   (padded to 16 h)
// ---------------------------------------------------------------------------
__global__ void k0_qproj(const float* __restrict__ w_kv,
                         const float* __restrict__ query,
                         float* __restrict__ qp) {
    int gid = blockIdx.x * blockDim.x + threadIdx.x;   // [0, 1024*16)
    if (gid >= kD * 16) return;
    int c = gid >> 4;
    int h = gid & 15;
    float acc = 0.0f;
    if (h < kH) {
        const float* wr = w_kv + (size_t)c * (2 * kP) + h * kHD;
        const float* qr = query + h * kHD;
        #pragma unroll 8
        for (int d = 0; d < kHD; ++d) acc += wr[d] * qr[d];
    }
    qp[c * 16 + h] = acc;
}

// ---------------------------------------------------------------------------
// K1: logits[b, t, h16] = (x[b, t, :] @ qp[:, h16]) / sqrt(HD)
//     WMMA f32 16x16x4: one wave per (b, 16-row tile of T). K loop = D/4.
//     A (16x4, M=x rows): lanes 0-15 hold M=lane, K={0,1}; lanes 16-31 K={2,3}
//     B (4x16, N=head):   lanes 0-15 hold N=lane, K={0,1}; lanes 16-31 K={2,3}
//     C (16x16): VGPR r -> row M = r + 8*(lane>=16), col N = lane&15
// ---------------------------------------------------------------------------
__global__ void k1_logits(const float* __restrict__ x,
                          const float* __restrict__ qp,
                          float* __restrict__ logits) {
    const int blk  = blockIdx.x;            // [0, B * T/16)
    const int b    = blk >> 8;              // T/16 = 256 tiles per batch
    const int mt   = blk & 255;
    const int lane = threadIdx.x & 31;
    const int half = lane >> 4;
    const int lr   = lane & 15;

    const float* xrow = x + ((size_t)b * kT + mt * 16 + lr) * kD;

    v8f acc = {};
    for (int k = 0; k < kD / 4; ++k) {
        const int kk = 4 * k + 2 * half;
        v2f a, bm;
        a.x  = xrow[kk];
        a.y  = xrow[kk + 1];
        bm.x = qp[kk * 16 + lr];
        bm.y = qp[(kk + 1) * 16 + lr];
        acc = __builtin_amdgcn_wmma_f32_16x16x4_f32(false, a, false, bm,
                                                    (short)0, acc, false, false);
    }
    const float scale = 0.08838834764831843f;   // 1/sqrt(128)
    #pragma unroll
    for (int r = 0; r < 8; ++r) {
        int row = mt * 16 + r + 8 * half;
        logits[((size_t)b * kT + row) * 16 + lr] = acc[r] * scale;
    }
}

// ---------------------------------------------------------------------------
// K2a: per (b, window, head): masked softmax stats (max, sum) over 64 slots.
//      stats[((b*W + w)*H + h)*2 + {0,1}] = {max, sum}
// ---------------------------------------------------------------------------
__global__ void k2a_stats(const float* __restrict__ logits,
                          float* __restrict__ stats) {
    __shared__ float red[64];
    const int bw = blockIdx.x;          // [0, B*W)
    const int b  = bw / kW;
    const int w  = bw % kW;
    const int t0 = w * kSTR;
    const int tt = threadIdx.x;         // [0, 64)
    const int t  = t0 + tt;
    const bool valid = (t < kT);
    const int tc = valid ? t : (kT - 1);

    for (int h = 0; h < kH; ++h) {
        float v = valid ? logits[((size_t)b * kT + tc) * 16 + h] : -INFINITY;
        red[tt] = v;
        __syncthreads();
        for (int s = 32; s > 0; s >>= 1) {
            if (tt < s) red[tt] = fmaxf(red[tt], red[tt + s]);
            __syncthreads();
        }
        float mx = red[0];
        __syncthreads();
        float e = valid ? __expf(v - mx) : 0.0f;
        red[tt] = e;
        __syncthreads();
        for (int s = 32; s > 0; s >>= 1) {
            if (tt < s) red[tt] += red[tt + s];
            __syncthreads();
        }
        if (tt == 0) {
            size_t si = (((size_t)bw) * kH + h) * 2;
            stats[si]     = mx;
            stats[si + 1] = red[0];
        }
        __syncthreads();
    }
}

// ---------------------------------------------------------------------------
// K2b: g[b, h16, t] = (1/W) * sum_{w contains t} exp(l - mx_w) / sum_w
//      Deterministic (no atomics); writes every element incl. zero pad rows.
// ---------------------------------------------------------------------------
__global__ void k2b_g(const float* __restrict__ logits,
                      const float* __restrict__ stats,
                      float* __restrict__ g) {
    int gid = blockIdx.x * blockDim.x + threadIdx.x;   // [0, B*16*T)
    if (gid >= kB * 16 * kT) return;
    const int t   = gid & (kT - 1);
    const int h16 = (gid >> 12) & 15;
    const int b   = gid >> 16;

    float acc = 0.0f;
    if (h16 < kH) {
        const float l = logits[((size_t)b * kT + t) * 16 + h16];
        const int w0 = t >> 5;                 // t / STRIDE
        #pragma unroll
        for (int j = 0; j < 2; ++j) {
            int w = w0 - j;
            if (w >= 0 && w < kW) {
                size_t si = ((((size_t)b * kW + w) * kH) + h16) * 2;
                float mx = stats[si];
                float sm = stats[si + 1];
                acc += __expf(l - mx) / sm;
            }
        }
        acc *= (1.0f / (float)kW);
    }
    g[gid] = acc;
}

// ---------------------------------------------------------------------------
// K3: partial y: ymp[kc][b][h16][c] = sum_{t in chunk kc} g[b,h16,t]*x[b,t,c]
//     WMMA f32 16x16x4: M=16 (8 heads + pad), K=512 per chunk, N=16-col tile.
//     One wave per (b, col tile, k-chunk)  -> 4096 independent waves.
// ---------------------------------------------------------------------------
__global__ void k3_y_partial(const float* __restrict__ x,
                             const float* __restrict__ g,
                             float* __restrict__ ymp) {
    const int blk  = blockIdx.x;        // [0, B * 64 * 8)
    const int b    = blk >> 9;
    const int r2   = blk & 511;
    const int ct   = r2 >> 3;           // column tile [0,64)
    const int kc   = r2 & 7;            // k chunk    [0,8)
    const int lane = threadIdx.x & 31;
    const int half = lane >> 4;
    const int lr   = lane & 15;

    const float* grow = g + ((size_t)b * 16 + lr) * kT + kc * kKCL;
    const float* xcol = x + ((size_t)b * kT + kc * kKCL) * kD + ct * 16 + lr;

    v8f acc = {};
    for (int k = 0; k < kKCL / 4; ++k) {
        const int kk = 4 * k + 2 * half;
        v2f a, bm;
        a.x  = grow[kk];
        a.y  = grow[kk + 1];
        bm.x = xcol[(size_t)kk * kD];
        bm.y = xcol[(size_t)(kk + 1) * kD];
        acc = __builtin_amdgcn_wmma_f32_16x16x4_f32(false, a, false, bm,
                                                    (short)0, acc, false, false);
    }
    #pragma unroll
    for (int r = 0; r < 8; ++r) {
        int h = r + 8 * half;
        ymp[(((size_t)kc * kB + b) * 16 + h) * kD + ct * 16 + lr] = acc[r];
    }
}

// ---------------------------------------------------------------------------
// K3r: ym[i] = sum_kc ymp[kc][i]   (fixed order -> deterministic)
// ---------------------------------------------------------------------------
__global__ void k3r_reduce(const float* __restrict__ ymp,
                           float* __restrict__ ym) {
    int gid = blockIdx.x * blockDim.x + threadIdx.x;   // [0, B*16*D)
    if (gid >= kB * 16 * kD) return;
    float s = 0.0f;
    #pragma unroll
    for (int kc = 0; kc < kKC; ++kc)
        s += ymp[(size_t)kc * kB * 16 * kD + gid];
    ym[gid] = s;
}

// ---------------------------------------------------------------------------
// K4: pooled[b, h*128+dh] = sum_c y[b, h, c] * w_kv[c, P + h*128+dh]
// ---------------------------------------------------------------------------
__global__ void k4_pooled(const float* __restrict__ ym,
                          const float* __restrict__ w_kv,
                          float* __restrict__ pooled) {
    const int bh = blockIdx.x;          // [0, B*H)
    const int b  = bh >> 3;
    const int h  = bh & 7;
    const int dh = threadIdx.x;         // [0, 128)
    const float* yr = ym + ((size_t)b * 16 + h) * kD;
    const float* wv = w_kv + kP + h * kHD + dh;
    float acc = 0.0f;
    for (int c = 0; c < kD; ++c)
        acc += yr[c] * wv[(size_t)c * (2 * kP)];
    pooled[(size_t)b * kP + h * kHD + dh] = acc;
}

// ---------------------------------------------------------------------------
// K5: out[b, o] = sum_p pooled[b, p] * w_out_w[o, p] + w_out_b[o]
// ---------------------------------------------------------------------------
__global__ void k5_out(const float* __restrict__ pooled,
                       const float* __restrict__ w_out_w,
                       const float* __restrict__ w_out_b,
                       float* __restrict__ out) {
    __shared__ float sp[kP];
    const int b = blockIdx.x;
    for (int p = threadIdx.x; p < kP; p += blockDim.x)
        sp[p] = pooled[(size_t)b * kP + p];
    __syncthreads();
    for (int o = threadIdx.x; o < kOUT; o += blockDim.x) {
        const float* wr = w_out_w + (size_t)o * kP;
        float acc = w_out_b[o];
        #pragma unroll 8
        for (int p = 0; p < kP; ++p) acc += sp[p] * wr[p];
        out[(size_t)b * kOUT + o] = acc;
    }
}

// ---------------------------------------------------------------------------
extern "C" void kernel_launch(void* const* d_in, const int* in_sizes, int n_in,
                              void* d_out, int out_size, void* d_ws, size_t ws_size,
                              hipStream_t stream) {
    const float* x       = (const float*)d_in[0];
    const float* w_kv    = (const float*)d_in[1];
    const float* query   = (const float*)d_in[2];
    const float* w_out_w = (const float*)d_in[3];
    const float* w_out_b = (const float*)d_in[4];
    float* out = (float*)d_out;

    // Workspace layout (floats), ~9 MB total
    float* qp     = (float*)d_ws;                       // 1024*16
    float* logits = qp + (size_t)kD * 16;               // B*T*16
    float* stats  = logits + (size_t)kB * kT * 16;      // B*W*H*2
    float* g      = stats + (size_t)kB * kW * kH * 2;   // B*16*T
    float* ymp    = g + (size_t)kB * 16 * kT;           // KC*B*16*D
    float* ym     = ymp + (size_t)kKC * kB * 16 * kD;   // B*16*D
    float* pooled = ym + (size_t)kB * 16 * kD;          // B*P

    // K0: q_proj (1024 x 16, padded)
    k0_qproj<<<(kD * 16 + 255) / 256, 256, 0, stream>>>(w_kv, query, qp);
    // K1: logits via WMMA (2048 waves; reads x once)
    k1_logits<<<kB * (kT / 16), 32, 0, stream>>>(x, qp, logits);
    // K2a: per-window softmax stats (max, sum)
    k2a_stats<<<kB * kW, 64, 0, stream>>>(logits, stats);
    // K2b: deterministic g (mean-over-windows folded in; pad rows zeroed)
    k2b_g<<<(kB * 16 * kT + 255) / 256, 256, 0, stream>>>(logits, stats, g);
    // K3: y partials via WMMA (4096 waves; reads x once)
    k3_y_partial<<<kB * 64 * kKC, 32, 0, stream>>>(x, g, ymp);
    // K3r: deterministic reduction of K partials
    k3r_reduce<<<(kB * 16 * kD + 255) / 256, 256, 0, stream>>>(ymp, ym);
    // K4: per-head pooled projection
    k4_pooled<<<kB * kH, kHD, 0, stream>>>(ym, w_kv, pooled);
    // K5: output projection + bias
    k5_out<<<kB, 256, 0, stream>>>(pooled, w_out_w, w_out_b, out);
}